// SortingCrossEntropyLoss_15805479650024
// MI455X (gfx1250) — compile-verified
//
#include <hip/hip_runtime.h>
#include <hip/hip_bf16.h>
#include <math.h>

#define N    2048
#define NCOL 8

typedef __attribute__((ext_vector_type(2))) float v2f;
typedef __attribute__((ext_vector_type(8))) float v8f;

// ---------------------------------------------------------------------------
// Kernel 1: per-column preprocessing.
//  For column c (one block per column):
//   - s[j], A_sum[j] (tau=1 so b[j]=A_sum[j])
//   - stable rank + exclusive event count  ->  lo/hi of the uniform GT row
//   - row max m_i and logZ_i of the NeuralSort logits -> off[i] = m_i + logZ_i
// ---------------------------------------------------------------------------
__global__ __launch_bounds__(1024)
void sce_prep(const float* __restrict__ logits,
              const int*   __restrict__ events,
              const float* __restrict__ durations,
              float* __restrict__ scol,
              float* __restrict__ bcol,
              float* __restrict__ offc,
              int*   __restrict__ locol,
              int*   __restrict__ hicol) {
  __shared__ float s_l[N];
  __shared__ float d_l[N];
  __shared__ int   e_l[N];
  __shared__ float b_l[N];

  const int c   = blockIdx.x;
  const int tid = threadIdx.x;

  for (int j = tid; j < N; j += 1024) {
    float sv = logits[j * NCOL + c];
    s_l[j] = sv;
    d_l[j] = durations[j * NCOL + c];
    e_l[j] = events[j * NCOL + c];
    scol[c * N + j] = sv;
  }
  __syncthreads();

  // A_sum, stable rank (argsort tie-break by index), exclusive event count
  for (int j = tid; j < N; j += 1024) {
    const float sj = s_l[j];
    const float dj = d_l[j];
    float asum = 0.0f;
    int rank = 0, ce = 0;
    for (int k = 0; k < N; ++k) {
      asum += fabsf(sj - s_l[k]);
      const float dk = d_l[k];
      const int before = (dk < dj) || ((dk == dj) && (k < j));
      rank += before;
      ce   += before & e_l[k];
    }
    b_l[j] = asum;                 // tau == 1
    bcol[c * N + j]  = asum;
    locol[c * N + j] = ce;                         // lo = exclusive event count
    hicol[c * N + j] = e_l[j] ? (rank + 1) : N;    // hi
  }
  __syncthreads();

  // Row softmax stats: score[i][j] = a_i*s_j - b_j,  a_i = n+1-2(i+1) = 2047-2i
  for (int i = tid; i < N; i += 1024) {
    const float a = (float)(2047 - 2 * i);
    float m = -3.4e38f;
    for (int k = 0; k < N; ++k) m = fmaxf(m, fmaf(a, s_l[k], -b_l[k]));
    float z = 0.0f;
    for (int k = 0; k < N; ++k) z += __expf(fmaf(a, s_l[k], -b_l[k]) - m);
    offc[c * N + i] = m + __logf(z);               // off = m + logZ
  }
}

// ---------------------------------------------------------------------------
// Kernel 2: fused BCE pass.
//  logp tile via one V_WMMA_F32_16X16X4_F32 per 16x16 tile:
//   A (16x4):  row m -> { a_m, off_m, 1, 0 }
//   B (4x16):  col n -> { s_n, -1, -b_n, 0 }
//   D[m][n] = a_m*s_n - off_m - b_n = logp[m][n]  (<= 0)
//  A layout (ISA 7.12.2, 32-bit 16x4): lanes 0-15 hold K=0,1 ; lanes 16-31 K=2,3.
//  B uses the matching half-range striping: V0 = K0|K2, V1 = K1|K3.
//  D: VGPR r, lanes 0-15 -> (M=r, N=lane); lanes 16-31 -> (M=r+8, N=lane-16).
//
//  Elementwise BCE is branch-free: one v_exp_f32 + one v_log_f32 per element,
//  with a 6-term expm1 Taylor (valid since the select only uses it for
//  uu in (-0.6932, 0]) instead of ocml's multi-dozen-op expm1f/log1pf.
// ---------------------------------------------------------------------------
__global__ __launch_bounds__(256)
void sce_bce(const float* __restrict__ scol,
             const float* __restrict__ bcol,
             const float* __restrict__ offc,
             const int*   __restrict__ locol,
             const int*   __restrict__ hicol,
             float*       __restrict__ partials) {
  const int c    = blockIdx.x >> 4;     // column
  const int tg   = blockIdx.x & 15;     // row-tile group (8 tiles per block)
  const int wave = threadIdx.x >> 5;
  const int lane = threadIdx.x & 31;
  const int half = lane >> 4;
  const int ln   = lane & 15;
  const int m0   = (tg * 8 + wave) * 16;

  const float* s   = scol  + c * N;
  const float* b   = bcol  + c * N;
  const float* off = offc  + c * N;
  const int*   lo  = locol + c * N;
  const int*   hi  = hicol + c * N;

  // A operand (loop invariant)
  v2f A;
  if (half == 0) {
    const int row = m0 + ln;
    A.x = (float)(2047 - 2 * row);  // K=0
    A.y = off[row];                 // K=1 (multiplies B row of -1)
  } else {
    A.x = 1.0f;                     // K=2 (multiplies -b_n)
    A.y = 0.0f;                     // K=3
  }

  // B operand: hoist the half-select into a per-lane base pointer + sign so
  // the inner-loop load is straight-line (no exec-mask branching).
  const float* pB = (half == 0) ? s : b;
  const float  bs = (half == 0) ? 1.0f : -1.0f;
  const float  by = (half == 0) ? -1.0f : 0.0f;

  // GT row metadata for the 8 rows this lane sees in D
  int   loR[8], hiR[8];
  float wR[8];
#pragma unroll
  for (int r = 0; r < 8; ++r) {
    const int row = m0 + r + 8 * half;
    const int l = lo[row], h = hi[row];
    loR[r] = l; hiR[r] = h;
    wR[r]  = 1.0f / (float)(h - l);
  }

  float acc = 0.0f;
  float vcur = bs * pB[ln];               // software-pipelined B.x
  for (int n0 = 0; n0 < N; n0 += 16) {
    const int jn = n0 + ln;
    float vnext = 0.0f;
    if (n0 + 16 < N) vnext = bs * pB[jn + 16];   // overlaps elementwise tail

    v2f B; B.x = vcur; B.y = by;
    v8f cz = {};
    v8f u = __builtin_amdgcn_wmma_f32_16x16x4_f32(
        /*neg_a=*/false, A, /*neg_b=*/false, B,
        /*c_mod=*/(short)0, cz, /*reuse_a=*/false, /*reuse_b=*/false);

#pragma unroll
    for (int r = 0; r < 8; ++r) {
      const float uu   = u[r];                       // logp (<= 0)
      const float logp = fmaxf(uu, -100.0f);
      const float p    = __expf(uu);
      // expm1 Taylor (6 terms), only selected when uu in (-0.6932, 0]
      const float em = uu * (1.0f + uu * (0.5f + uu * (0.16666667f +
                       uu * (4.1666668e-2f + uu * (8.3333338e-3f +
                       uu * 1.3888889e-3f)))));
      const float arg = (uu > -0.6931472f) ? -em : (1.0f - p);
      const float l1m = fmaxf(__logf(arg), -100.0f);
      const float y   = (jn >= loR[r] && jn < hiR[r]) ? wR[r] : 0.0f;
      acc += l1m + y * (logp - l1m);
    }
    vcur = vnext;
  }

  // wave32 reduction, then one deterministic partial per block
#pragma unroll
  for (int o = 16; o > 0; o >>= 1) acc += __shfl_xor(acc, o, 32);
  __shared__ float red[8];
  if (lane == 0) red[wave] = acc;
  __syncthreads();
  if (threadIdx.x == 0) {
    float t = 0.0f;
#pragma unroll
    for (int w = 0; w < 8; ++w) t += red[w];
    partials[blockIdx.x] = t;
  }
}

// ---------------------------------------------------------------------------
// Kernel 3: deterministic finalize (masked mean over columns).
// ---------------------------------------------------------------------------
__global__ void sce_final(const float* __restrict__ partials,
                          float* __restrict__ out) {
  if (threadIdx.x == 0 && blockIdx.x == 0) {
    float total = 0.0f;
    int cnt = 0;
    for (int c = 0; c < NCOL; ++c) {
      float sum = 0.0f;
      for (int t = 0; t < 16; ++t) sum += partials[c * 16 + t];
      const float loss = -sum * (1.0f / 4194304.0f);   // / n^2
      if (loss > 0.0f) { total += loss; cnt++; }
    }
    out[0] = total / (float)(cnt > 0 ? cnt : 1);
  }
}

extern "C" void kernel_launch(void* const* d_in, const int* in_sizes, int n_in,
                              void* d_out, int out_size, void* d_ws, size_t ws_size,
                              hipStream_t stream) {
  (void)in_sizes; (void)n_in; (void)out_size; (void)ws_size;
  const float* logits    = (const float*)d_in[0];
  const int*   events    = (const int*)  d_in[1];
  const float* durations = (const float*)d_in[2];

  // workspace carve-out: 5 arrays of NCOL*N + 128 block partials (~321 KB)
  float* scol     = (float*)d_ws;
  float* bcol     = scol + NCOL * N;
  float* offc     = bcol + NCOL * N;
  int*   locol    = (int*)(offc + NCOL * N);
  int*   hicol    = locol + NCOL * N;
  float* partials = (float*)(hicol + NCOL * N);

  sce_prep<<<NCOL, 1024, 0, stream>>>(logits, events, durations,
                                      scol, bcol, offc, locol, hicol);
  sce_bce<<<NCOL * 16, 256, 0, stream>>>(scol, bcol, offc, locol, hicol, partials);
  sce_final<<<1, 32, 0, stream>>>(partials, (float*)d_out);
}